// IOVPreTrainedEmbeddings_23252952940731
// MI455X (gfx1250) — compile-verified
//
#include <hip/hip_runtime.h>
#include <hip/hip_bf16.h>
#include <stdint.h>

// IOVPreTrainedEmbeddings: masked dual-table embedding gather.
//   out[t, :] = (oov_map[x[t]] >= 0) ? oov_embed[oov_map[x[t]], :] : w2v[x[t], :]
// D = 300 floats = 1200 bytes = 75 x 16B chunks per row (16B-aligned everywhere).
//
// Pure data movement (0 FLOPs): ~315 MB total traffic -> ~13.5 us roofline at
// 23.3 TB/s. Strategy: one wave32 per token, copy the row global->LDS->global
// with the CDNA5 async data mover (b128, ASYNCcnt-tracked, VGPR-free), and
// stream the output with th:TH_STORE_NT so the 126 MB of embedding tables
// stay resident in the 192 MB L2.

#define DIM          300
#define CHUNKS       75            // 16B chunks per row
#define WAVES_PER_BLOCK 8
#define BLOCK_THREADS   (WAVES_PER_BLOCK * 32)

__global__ __launch_bounds__(BLOCK_THREADS)
void IOVPreTrainedEmbeddings_gather_kernel(const int* __restrict__ x,
                                           const float* __restrict__ w2v,
                                           const float* __restrict__ oov_embed,
                                           const int* __restrict__ oov_map,
                                           float* __restrict__ out,
                                           int n_tokens) {
    // Per-wave 1200B staging slice; 16B aligned for b128 LDS writes.
    __shared__ __align__(16) float lds_buf[WAVES_PER_BLOCK * DIM];

    const int wave = threadIdx.x >> 5;
    const int lane = threadIdx.x & 31;
    const int t    = blockIdx.x * WAVES_PER_BLOCK + wave;   // token index
    if (t >= n_tokens) return;

    // Uniform-per-wave index loads (all 32 lanes hit the same address ->
    // single coalesced request, then per-lane pointer select).
    const int id  = x[t];
    const int row = oov_map[id];

    const float* src = (row >= 0) ? (oov_embed + (size_t)row * DIM)
                                  : (w2v       + (size_t)id  * DIM);
    float* dst = out + (size_t)t * DIM;

    // Wave-relative LDS byte address (low 32 bits of the generic pointer).
    const unsigned lds_base = (unsigned)(uintptr_t)(&lds_buf[wave * DIM]);

    // ---- global -> LDS: 75 x b128 async chunks (3 rounds of 32 lanes) ----
    #pragma unroll
    for (int i = 0; i < 3; ++i) {
        const int c = lane + 32 * i;
        if (c < CHUNKS) {
            const float* g = src + c * 4;
            const unsigned l = lds_base + c * 16;
            asm volatile("global_load_async_to_lds_b128 %0, %1, off"
                         :: "v"(l), "v"(g)
                         : "memory");
        }
    }

    // Wait for the async loads to land in LDS before reading it back out.
    asm volatile("s_wait_asynccnt 0x0" ::: "memory");

    // ---- LDS -> global: non-temporal streamed output (write-once data) ----
    #pragma unroll
    for (int i = 0; i < 3; ++i) {
        const int c = lane + 32 * i;
        if (c < CHUNKS) {
            float* g = dst + c * 4;
            const unsigned l = lds_base + c * 16;
            asm volatile("global_store_async_from_lds_b128 %0, %1, off th:TH_STORE_NT"
                         :: "v"(g), "v"(l)
                         : "memory");
        }
    }
    // S_ENDPGM performs an implicit wait-idle, but be explicit: the LDS slice
    // must not be considered free until the async stores have consumed it.
    asm volatile("s_wait_asynccnt 0x0" ::: "memory");
}

extern "C" void kernel_launch(void* const* d_in, const int* in_sizes, int n_in,
                              void* d_out, int out_size, void* d_ws, size_t ws_size,
                              hipStream_t stream) {
    // setup_inputs() order: x [B*S], w2v [VOCAB*DIM], oov_embed [N_OOV*DIM],
    //                       oov_map [VOCAB]
    const int*   x        = (const int*)d_in[0];
    const float* w2v      = (const float*)d_in[1];
    const float* oov      = (const float*)d_in[2];
    const int*   oov_map  = (const int*)d_in[3];
    float*       out      = (float*)d_out;

    const int n_tokens = in_sizes[0];                       // 131072
    const int blocks   = (n_tokens + WAVES_PER_BLOCK - 1) / WAVES_PER_BLOCK;

    IOVPreTrainedEmbeddings_gather_kernel<<<blocks, BLOCK_THREADS, 0, stream>>>(
        x, w2v, oov, oov_map, out, n_tokens);
}